// BasicGATModel_57561151701654
// MI455X (gfx1250) — compile-verified
//
#include <hip/hip_runtime.h>

typedef __attribute__((ext_vector_type(2))) float v2f;
typedef __attribute__((ext_vector_type(8))) float v8f;

#define NEG_SLOPE 0.2f

// ---------------------------------------------------------------------------
// GEMM  Y[M,C] = X[M,K] @ W[K,C]   (fp32, V_WMMA_F32_16X16X4_F32, wave32)
// Block = 128 threads = 4 waves; one 16-row tile per block; waves own col tiles.
// K <= 128.
// A layout : lane l, vgpr g -> M = l%16, K = 2*(l/16)+g
// B layout : lane l, vgpr g -> N = l%16, K = 2*(l/16)+g
// C/D      : lane l, vgpr g -> M = g + 8*(l/16), N = l%16
// ---------------------------------------------------------------------------
__global__ void gemm_wmma_f32(const float* __restrict__ X, const float* __restrict__ W,
                              float* __restrict__ Y, int M, int K, int C) {
  __shared__ float As[16 * 128];
  const int m0 = blockIdx.x * 16;
  const int tid = threadIdx.x;

  // cooperative stage of the 16 x K A-tile into LDS (row-clamped for tail tile)
  for (int idx = tid; idx < 16 * K; idx += 128) {
    int r = idx / K, c = idx - r * K;
    int gr = m0 + r;
    if (gr >= M) gr = M - 1;
    As[idx] = X[(long)gr * K + c];
  }
  __syncthreads();

  const int wave  = tid >> 5;
  const int lane  = tid & 31;
  const int lhalf = lane >> 4;   // 0 or 1
  const int lmod  = lane & 15;
  const int ntiles = C >> 4;

  for (int nt = wave; nt < ntiles; nt += 4) {
    const int n0 = nt * 16;
    v8f acc = {0.f, 0.f, 0.f, 0.f, 0.f, 0.f, 0.f, 0.f};
    for (int k0 = 0; k0 < K; k0 += 4) {
      const int ka = k0 + 2 * lhalf;
      v2f a, b;
      a.x = As[lmod * K + ka];
      a.y = As[lmod * K + ka + 1];
      b.x = W[(long)ka * C + n0 + lmod];
      b.y = W[(long)(ka + 1) * C + n0 + lmod];
      acc = __builtin_amdgcn_wmma_f32_16x16x4_f32(
          /*neg_a=*/false, a, /*neg_b=*/false, b,
          /*c_mod=*/(short)0, acc, /*reuse_a=*/false, /*reuse_b=*/false);
    }
    for (int g = 0; g < 8; ++g) {
      int m = m0 + g + 8 * lhalf;
      if (m < M) Y[(long)m * C + n0 + lmod] = acc[g];
    }
  }
}

// ---------------------------------------------------------------------------
// el[n,h] = sum_f hp[n,h,f]*al[h,f] ; er likewise
// ---------------------------------------------------------------------------
__global__ void attn_dots(const float* __restrict__ hp, const float* __restrict__ al,
                          const float* __restrict__ ar, float* __restrict__ el,
                          float* __restrict__ er, int Nn, int H, int F) {
  int t = blockIdx.x * blockDim.x + threadIdx.x;
  if (t >= Nn * H) return;
  int n = t / H, h = t - n * H;
  const float* row = hp + (long)n * H * F + (long)h * F;
  const float* a0  = al + h * F;
  const float* a1  = ar + h * F;
  float sl = 0.f, sr = 0.f;
  for (int f = 0; f < F; ++f) {
    float v = row[f];
    sl += v * a0[f];
    sr += v * a1[f];
  }
  el[t] = sl;
  er[t] = sr;
}

__global__ void fill_u32_k(unsigned* __restrict__ p, unsigned v, int n) {
  int t = blockIdx.x * blockDim.x + threadIdx.x;
  if (t < n) p[t] = v;
}

__global__ void init_bias_k(float* __restrict__ out, const float* __restrict__ b,
                            int Nn, int C) {
  int t = blockIdx.x * blockDim.x + threadIdx.x;
  if (t < Nn * C) out[t] = b[t % C];
}

// monotone float <-> uint map for atomic max over signed floats
__device__ __forceinline__ unsigned enc_f(float x) {
  unsigned u = __float_as_uint(x);
  return (u & 0x80000000u) ? ~u : (u | 0x80000000u);
}
__device__ __forceinline__ float dec_f(unsigned u) {
  return __uint_as_float((u & 0x80000000u) ? (u & 0x7FFFFFFFu) : ~u);
}

// ---------------------------------------------------------------------------
// pass 1: e = leaky_relu(el[src]+er[dst]); segment max into mEnc[dst]
// ---------------------------------------------------------------------------
__global__ void edge_score_max(const int* __restrict__ src, const int* __restrict__ dst,
                               const float* __restrict__ el, const float* __restrict__ er,
                               float* __restrict__ ebuf, unsigned* __restrict__ mEnc,
                               int E_, int H) {
  int t = blockIdx.x * blockDim.x + threadIdx.x;
  if (t >= E_ * H) return;
  int e = t / H, h = t - e * H;
  int s = src[e], d = dst[e];
  float z = el[s * H + h] + er[d * H + h];
  float v = (z > 0.f) ? z : NEG_SLOPE * z;
  ebuf[t] = v;
  atomicMax(&mEnc[d * H + h], enc_f(v));
}

// ---------------------------------------------------------------------------
// pass 2: a = exp(e - m[dst]); segment sum into denom[dst]  (ebuf updated in place)
// ---------------------------------------------------------------------------
__global__ void edge_exp_sum(const int* __restrict__ dst, float* __restrict__ ebuf,
                             const unsigned* __restrict__ mEnc, float* __restrict__ denom,
                             int E_, int H) {
  int t = blockIdx.x * blockDim.x + threadIdx.x;
  if (t >= E_ * H) return;
  int e = t / H, h = t - e * H;
  int d = dst[e];
  float a = __expf(ebuf[t] - dec_f(mEnc[d * H + h]));
  ebuf[t] = a;
  atomicAdd(&denom[d * H + h], a);
}

// ---------------------------------------------------------------------------
// pass 3: out[dst,h,f] += (a/denom[dst,h]) * hp[src,h,f]
// one thread per (edge, h, f)
// ---------------------------------------------------------------------------
__global__ void edge_aggregate(const int* __restrict__ src, const int* __restrict__ dst,
                               const float* __restrict__ abuf, const float* __restrict__ denom,
                               const float* __restrict__ hp, float* __restrict__ out,
                               int E_, int H, int F) {
  long t = (long)blockIdx.x * blockDim.x + threadIdx.x;
  const int HF = H * F;
  long total = (long)E_ * HF;
  if (t >= total) return;
  int e = (int)(t / HF);
  int r = (int)(t - (long)e * HF);
  int h = r / F;
  int s = src[e], d = dst[e];
  float alpha = abuf[(long)e * H + h] / denom[d * H + h];
  atomicAdd(&out[(long)d * HF + r], alpha * hp[(long)s * HF + r]);
}

// ---------------------------------------------------------------------------
extern "C" void kernel_launch(void* const* d_in, const int* in_sizes, int n_in,
                              void* d_out, int out_size, void* d_ws, size_t ws_size,
                              hipStream_t stream) {
  const float* x   = (const float*)d_in[0];
  const float* W1  = (const float*)d_in[1];
  const float* al1 = (const float*)d_in[2];
  const float* ar1 = (const float*)d_in[3];
  const float* b1  = (const float*)d_in[4];
  const float* W2  = (const float*)d_in[5];
  const float* al2 = (const float*)d_in[6];
  const float* ar2 = (const float*)d_in[7];
  const float* b2  = (const float*)d_in[8];
  const int*   src = (const int*)d_in[9];
  const int*   dst = (const int*)d_in[10];

  const int DIN = 128, H1 = 4, F1 = 32, HF1 = 128, C2 = 64;
  const int Nn = in_sizes[0] / DIN;
  const int E_ = in_sizes[9];

  char* ws = (char*)d_ws;
  size_t off = 0;
  auto carve = [&](size_t bytes) -> void* {
    void* p = ws + off;
    off = (off + bytes + 255) & ~(size_t)255;
    return p;
  };
  float*    hp1  = (float*)carve((size_t)Nn * HF1 * 4);   // also reused as hp2
  float*    h1   = (float*)carve((size_t)Nn * HF1 * 4);
  float*    el   = (float*)carve((size_t)Nn * H1 * 4);
  float*    er   = (float*)carve((size_t)Nn * H1 * 4);
  unsigned* mEnc = (unsigned*)carve((size_t)Nn * H1 * 4);
  float*    den  = (float*)carve((size_t)Nn * H1 * 4);
  float*    ebuf = (float*)carve((size_t)E_ * H1 * 4);
  float*    hp2  = hp1;   // layer-2 reuse (Nn*64 <= Nn*128)
  float*    out  = (float*)d_out;

  const int rowTiles = (Nn + 15) / 16;
  const int nh1 = Nn * H1;
  const int eh1 = E_ * H1;

  // ---------------- layer 1 : D_IN=128 -> 4 heads x 32 ----------------
  gemm_wmma_f32<<<rowTiles, 128, 0, stream>>>(x, W1, hp1, Nn, DIN, HF1);
  fill_u32_k<<<(nh1 + 255) / 256, 256, 0, stream>>>(mEnc, 0u, nh1);
  fill_u32_k<<<(nh1 + 255) / 256, 256, 0, stream>>>((unsigned*)den, 0u, nh1);
  attn_dots<<<(nh1 + 255) / 256, 256, 0, stream>>>(hp1, al1, ar1, el, er, Nn, H1, F1);
  init_bias_k<<<(Nn * HF1 + 255) / 256, 256, 0, stream>>>(h1, b1, Nn, HF1);
  edge_score_max<<<(eh1 + 255) / 256, 256, 0, stream>>>(src, dst, el, er, ebuf, mEnc, E_, H1);
  edge_exp_sum<<<(eh1 + 255) / 256, 256, 0, stream>>>(dst, ebuf, mEnc, den, E_, H1);
  {
    long tot = (long)E_ * HF1;
    int blocks = (int)((tot + 255) / 256);
    edge_aggregate<<<blocks, 256, 0, stream>>>(src, dst, ebuf, den, hp1, h1, E_, H1, F1);
  }

  // ---------------- layer 2 : 128 -> 1 head x 64 ----------------
  gemm_wmma_f32<<<rowTiles, 128, 0, stream>>>(h1, W2, hp2, Nn, HF1, C2);
  fill_u32_k<<<(Nn + 255) / 256, 256, 0, stream>>>(mEnc, 0u, Nn);
  fill_u32_k<<<(Nn + 255) / 256, 256, 0, stream>>>((unsigned*)den, 0u, Nn);
  attn_dots<<<(Nn + 255) / 256, 256, 0, stream>>>(hp2, al2, ar2, el, er, Nn, 1, C2);
  init_bias_k<<<(Nn * C2 + 255) / 256, 256, 0, stream>>>(out, b2, Nn, C2);
  edge_score_max<<<(E_ + 255) / 256, 256, 0, stream>>>(src, dst, el, er, ebuf, mEnc, E_, 1);
  edge_exp_sum<<<(E_ + 255) / 256, 256, 0, stream>>>(dst, ebuf, mEnc, den, E_, 1);
  {
    long tot = (long)E_ * C2;
    int blocks = (int)((tot + 255) / 256);
    edge_aggregate<<<blocks, 256, 0, stream>>>(src, dst, ebuf, den, hp2, out, E_, 1, C2);
  }
}